// mGNNBase_54795192762697
// MI455X (gfx1250) — compile-verified
//
#include <hip/hip_runtime.h>
#include <hip/hip_bf16.h>

#define NBLK   3
#define BATCH  8
#define NA     8
#define NF     512
#define NH     128
#define DMODEL 32
#define DHID   256

typedef __attribute__((ext_vector_type(2))) float v2f;
typedef __attribute__((ext_vector_type(8))) float v8f;

// ---------------------------------------------------------------------------
// Generic batched GEMM via V_WMMA_F32_16X16X4_F32 (f32, keeps ref precision).
// out[b] = act(concat(A0[b],A1[b]) @ W + bias) (+ resid[b])
// W shared across batch, row-major [K,N]. N multiple of 32 (two 16x16 tiles
// per wave sharing the A fragment). TAIL=true handles K<4 (encoder K=3) with
// value-masking only (clamped addresses, no EXEC divergence).
// Clean path requires K%4==0 and ksplit%4==0 (true for all call sites).
// ---------------------------------------------------------------------------
template <bool TAIL>
__global__ __launch_bounds__(128)
void gemm_wmma_kernel(const float* __restrict__ A0,
                      const float* __restrict__ A1,
                      const float* __restrict__ W,
                      const float* __restrict__ bias,
                      const float* __restrict__ resid,
                      float* __restrict__ out,
                      int M, int N, int K, int ksplit, int relu) {
  const int tilesN = N >> 5;                 // 32-wide strips
  const int tilesM = (M + 15) >> 4;
  const int tile = blockIdx.x * (blockDim.x >> 5) + (threadIdx.x >> 5);
  if (tile >= tilesM * tilesN) return;       // whole-wave exit; no LDS here
  const int tm = tile / tilesN;
  const int tn = tile - tm * tilesN;
  const int lane = threadIdx.x & 31;
  const int half = lane >> 4;                // A/B frag: lanes 0-15 K0/K1, 16-31 K2/K3
  const int l16  = lane & 15;
  const int b = blockIdx.y;

  const bool hasA1 = (A1 != nullptr);
  const int lda0 = hasA1 ? ksplit : K;
  const int lda1 = K - ksplit;

  int arow = tm * 16 + l16;
  if (arow >= M) arow = M - 1;               // clamp loads; stores masked below
  const int col = tn * 32 + l16;

  const float* Ar0 = A0 + ((size_t)b * M + arow) * lda0;
  const float* Wc0 = W + col;
  const float* Wc1 = W + col + 16;
  const int ka = half * 2;

  v8f c0 = {}, c1 = {};
  if (TAIL) {
    // Single padded K-step (K in [1,4)). Mask values, not EXEC.
    const int k0c = (ka     < K) ? ka       : (K - 1);
    const int k1c = (ka + 1 < K) ? (ka + 1) : (K - 1);
    v2f a, b0, b1;
    float ax = Ar0[k0c],              ay = Ar0[k1c];
    float bx0 = Wc0[(size_t)k0c * N], by0 = Wc0[(size_t)k1c * N];
    float bx1 = Wc1[(size_t)k0c * N], by1 = Wc1[(size_t)k1c * N];
    a.x  = (ka     < K) ? ax  : 0.0f;  a.y  = (ka + 1 < K) ? ay  : 0.0f;
    b0.x = (ka     < K) ? bx0 : 0.0f;  b0.y = (ka + 1 < K) ? by0 : 0.0f;
    b1.x = (ka     < K) ? bx1 : 0.0f;  b1.y = (ka + 1 < K) ? by1 : 0.0f;
    c0 = __builtin_amdgcn_wmma_f32_16x16x4_f32(false, a, false, b0, (short)0, c0, false, false);
    c1 = __builtin_amdgcn_wmma_f32_16x16x4_f32(false, a, false, b1, (short)0, c1, false, false);
  } else {
    const int klim0 = hasA1 ? ksplit : K;
    int k0 = 0;
#pragma unroll 2
    for (; k0 < klim0; k0 += 4) {            // A0 portion: contiguous b64 A load
      v2f a = *(const v2f*)(Ar0 + k0 + ka);
      v2f b0, b1;
      b0.x = Wc0[(size_t)(k0 + ka) * N];  b0.y = Wc0[(size_t)(k0 + ka + 1) * N];
      b1.x = Wc1[(size_t)(k0 + ka) * N];  b1.y = Wc1[(size_t)(k0 + ka + 1) * N];
      c0 = __builtin_amdgcn_wmma_f32_16x16x4_f32(false, a, false, b0, (short)0, c0, false, false);
      c1 = __builtin_amdgcn_wmma_f32_16x16x4_f32(false, a, false, b1, (short)0, c1, false, false);
    }
    if (hasA1) {
      const float* Ar1 = A1 + ((size_t)b * M + arow) * lda1;
#pragma unroll 2
      for (; k0 < K; k0 += 4) {              // A1 (concat) portion
        v2f a = *(const v2f*)(Ar1 + (k0 - klim0) + ka);
        v2f b0, b1;
        b0.x = Wc0[(size_t)(k0 + ka) * N];  b0.y = Wc0[(size_t)(k0 + ka + 1) * N];
        b1.x = Wc1[(size_t)(k0 + ka) * N];  b1.y = Wc1[(size_t)(k0 + ka + 1) * N];
        c0 = __builtin_amdgcn_wmma_f32_16x16x4_f32(false, a, false, b0, (short)0, c0, false, false);
        c1 = __builtin_amdgcn_wmma_f32_16x16x4_f32(false, a, false, b1, (short)0, c1, false, false);
      }
    }
  }

  const float bv0 = (bias != nullptr) ? bias[col]      : 0.0f;
  const float bv1 = (bias != nullptr) ? bias[col + 16] : 0.0f;
  float* outB = out + (size_t)b * M * N;
  const float* resB = (resid != nullptr) ? (resid + (size_t)b * M * N) : nullptr;
#pragma unroll
  for (int r = 0; r < 8; ++r) {
    const int orow = tm * 16 + r + half * 8; // C/D layout: M = r + 8*half
    if (orow < M) {
      float v0 = c0[r] + bv0;
      float v1 = c1[r] + bv1;
      if (relu) { v0 = fmaxf(v0, 0.0f); v1 = fmaxf(v1, 0.0f); }
      if (resB) {
        v0 += resB[(size_t)orow * N + col];
        v1 += resB[(size_t)orow * N + col + 16];
      }
      outB[(size_t)orow * N + col]      = v0;
      outB[(size_t)orow * N + col + 16] = v1;
    }
  }
}

// ---------------------------------------------------------------------------
// wave32 reductions
// ---------------------------------------------------------------------------
__device__ __forceinline__ float wave_reduce_sum(float x) {
  x += __shfl_xor(x, 1);  x += __shfl_xor(x, 2);  x += __shfl_xor(x, 4);
  x += __shfl_xor(x, 8);  x += __shfl_xor(x, 16);
  return x;
}
__device__ __forceinline__ float wave_reduce_max(float x) {
  x = fmaxf(x, __shfl_xor(x, 1));  x = fmaxf(x, __shfl_xor(x, 2));
  x = fmaxf(x, __shfl_xor(x, 4));  x = fmaxf(x, __shfl_xor(x, 8));
  x = fmaxf(x, __shfl_xor(x, 16));
  return x;
}

// ---------------------------------------------------------------------------
// Intra self-attention: one wave per query row. agg = softmax(q k^T) @ v
// ---------------------------------------------------------------------------
__global__ __launch_bounds__(128)
void attn_softmax_kernel(const float* __restrict__ q,
                         const float* __restrict__ k,
                         const float* __restrict__ v,
                         float* __restrict__ agg, int N) {
  __shared__ float srow[4][512];
  const int b = blockIdx.y;
  const int wave = threadIdx.x >> 5;
  const int lane = threadIdx.x & 31;
  int i = blockIdx.x * 4 + wave;
  if (i >= N) i = N - 1;                     // benign duplicate (same values)
  const float* qB = q + ((size_t)b * N + i) * DMODEL;
  const float* kB = k + (size_t)b * N * DMODEL;
  const float* vB = v + (size_t)b * N * DMODEL;
  const float qreg = qB[lane];               // d=32 == wave32: one dim per lane

  for (int j = 0; j < N; ++j) {
    float p = qreg * kB[(size_t)j * DMODEL + lane];   // coalesced
    p = wave_reduce_sum(p);
    if (lane == 0) srow[wave][j] = p;
  }
  __syncthreads();

  float m = -3.402823466e38f;
  for (int t = lane; t < N; t += 32) m = fmaxf(m, srow[wave][t]);
  m = wave_reduce_max(m);
  float ssum = 0.0f;
  for (int t = lane; t < N; t += 32) {
    float e = expf(srow[wave][t] - m);
    srow[wave][t] = e;
    ssum += e;
  }
  ssum = wave_reduce_sum(ssum);
  const float rinv = 1.0f / ssum;
  __syncthreads();

  float acc = 0.0f;
  for (int j = 0; j < N; ++j) acc += srow[wave][j] * vB[(size_t)j * DMODEL + lane];
  agg[((size_t)b * N + i) * DMODEL + lane] = acc * rinv;
}

// ---------------------------------------------------------------------------
// Inter edge-MLP + softmax + aggregate. One wave per query i.
// s_ij = relu(hq_i + hk_j + dis_ij*w_d + eb1) . eW2 + eb2 ; e = softmax_j(s)
// agg_i = e @ yv.  hq/wd/eb1/eW2 register-resident (8 f32/lane); hk streamed.
// ---------------------------------------------------------------------------
__global__ __launch_bounds__(128)
void inter_edge_kernel(const float* __restrict__ hq,
                       const float* __restrict__ hk,
                       const float* __restrict__ dis,
                       const float* __restrict__ wd,
                       const float* __restrict__ eb1,
                       const float* __restrict__ ew2,
                       const float* __restrict__ eb2p,
                       const float* __restrict__ yv,
                       float* __restrict__ agg,
                       float* __restrict__ e_out,
                       int Nx, int Ny) {
  __shared__ float srow[4][512];
  const int b = blockIdx.y;
  const int wave = threadIdx.x >> 5;
  const int lane = threadIdx.x & 31;
  int i = blockIdx.x * 4 + wave;
  if (i >= Nx) i = Nx - 1;
  const float* hqi  = hq  + ((size_t)b * Nx + i) * DHID;
  const float* hkB  = hk  + (size_t)b * Ny * DHID;
  const float* disB = dis + ((size_t)b * Nx + i) * Ny;

  float hqr[8], wdr[8], ebr[8], ewr[8];
#pragma unroll
  for (int t = 0; t < 8; ++t) {
    const int cidx = lane + 32 * t;
    hqr[t] = hqi[cidx];  wdr[t] = wd[cidx];
    ebr[t] = eb1[cidx];  ewr[t] = ew2[cidx];
  }
  const float eb2v = eb2p[0];

  for (int j = 0; j < Ny; ++j) {
    const float dv = disB[j];
    const float* hkj = hkB + (size_t)j * DHID;
    float acc = 0.0f;
#pragma unroll
    for (int t = 0; t < 8; ++t) {
      float h = hqr[t] + hkj[lane + 32 * t] + dv * wdr[t] + ebr[t];
      h = fmaxf(h, 0.0f);
      acc = fmaf(h, ewr[t], acc);
    }
    acc = wave_reduce_sum(acc);
    if (lane == 0) srow[wave][j] = acc + eb2v;
  }
  __syncthreads();

  float m = -3.402823466e38f;
  for (int t = lane; t < Ny; t += 32) m = fmaxf(m, srow[wave][t]);
  m = wave_reduce_max(m);
  float ssum = 0.0f;
  for (int t = lane; t < Ny; t += 32) {
    float e = expf(srow[wave][t] - m);
    srow[wave][t] = e;
    ssum += e;
  }
  ssum = wave_reduce_sum(ssum);
  const float rinv = 1.0f / ssum;
  __syncthreads();

  if (e_out != nullptr) {
    float* eo = e_out + ((size_t)b * Nx + i) * Ny;
    for (int t = lane; t < Ny; t += 32) eo[t] = srow[wave][t] * rinv;
  }
  const float* yvB = yv + (size_t)b * Ny * DMODEL;
  float acc2 = 0.0f;
  for (int j = 0; j < Ny; ++j) acc2 += srow[wave][j] * yvB[(size_t)j * DMODEL + lane];
  agg[((size_t)b * Nx + i) * DMODEL + lane] = acc2 * rinv;
}

// ---------------------------------------------------------------------------
// Orchestration
// ---------------------------------------------------------------------------
extern "C" void kernel_launch(void* const* d_in, const int* in_sizes, int n_in,
                              void* d_out, int out_size, void* d_ws, size_t ws_size,
                              hipStream_t stream) {
  (void)in_sizes; (void)n_in; (void)out_size; (void)ws_size;
  const float* agent_pos    = (const float*)d_in[0];
  const float* frontier_pos = (const float*)d_in[1];
  const float* past_agent   = (const float*)d_in[2];
  const float* past_goal    = (const float*)d_in[3];
  const float* dis_rp = (const float*)d_in[4];
  const float* dis_fp = (const float*)d_in[5];
  const float* dis_rf = (const float*)d_in[6];
  const float* ni_W1 = (const float*)d_in[7];
  const float* ni_b1 = (const float*)d_in[8];
  const float* ni_W2 = (const float*)d_in[9];
  const float* ni_b2 = (const float*)d_in[10];
  const float* Wq = (const float*)d_in[11];
  const float* bq = (const float*)d_in[12];
  const float* Wk = (const float*)d_in[13];
  const float* bk = (const float*)d_in[14];
  const float* Wv = (const float*)d_in[15];
  const float* bv = (const float*)d_in[16];
  const float* nW1 = (const float*)d_in[17];
  const float* nb1 = (const float*)d_in[18];
  const float* nW2 = (const float*)d_in[19];
  const float* nb2 = (const float*)d_in[20];
  const float* eW1 = (const float*)d_in[21];
  const float* eb1 = (const float*)d_in[22];
  const float* eW2 = (const float*)d_in[23];
  const float* eb2 = (const float*)d_in[24];

  float* ws = (float*)d_ws;
  size_t off = 0;
  auto bump = [&](size_t n) { float* p = ws + off; off += n; return p; };
  float* robot    = bump((size_t)BATCH * NA * DMODEL);
  float* frontier = bump((size_t)BATCH * NF * DMODEL);
  float* rh       = bump((size_t)BATCH * NH * DMODEL);
  float* fh       = bump((size_t)BATCH * NH * DMODEL);
  float* qb       = bump((size_t)BATCH * NF * DMODEL);
  float* kb       = bump((size_t)BATCH * NF * DMODEL);
  float* vvb      = bump((size_t)BATCH * NF * DMODEL);
  float* aggb     = bump((size_t)BATCH * NF * DMODEL);
  float* hid      = bump((size_t)BATCH * NF * DHID);
  float* hqb      = bump((size_t)BATCH * NF * DHID);
  float* hkb      = bump((size_t)BATCH * NF * DHID);

  auto gemm = [&](const float* A0, const float* A1, const float* W,
                  const float* bias, const float* resid, float* out,
                  int M, int N, int K, int ksplit, int relu) {
    int tiles = ((M + 15) / 16) * (N / 32);
    dim3 grid((tiles + 3) / 4, BATCH);
    if (K & 3)
      gemm_wmma_kernel<true><<<grid, 128, 0, stream>>>(A0, A1, W, bias, resid,
                                                       out, M, N, K, ksplit, relu);
    else
      gemm_wmma_kernel<false><<<grid, 128, 0, stream>>>(A0, A1, W, bias, resid,
                                                        out, M, N, K, ksplit, relu);
  };

  // node encoder: relu(pos@W1+b1)@W2+b2
  auto encode = [&](const float* pos, float* dst, int N) {
    gemm(pos, nullptr, ni_W1, ni_b1, nullptr, hid, N, DHID, 3, 0, 1);
    gemm(hid, nullptr, ni_W2, ni_b2, nullptr, dst, N, DMODEL, DHID, 0, 0);
  };
  encode(agent_pos,    robot,    NA);
  encode(frontier_pos, frontier, NF);
  encode(past_agent,   rh,       NH);
  encode(past_goal,    fh,       NH);

  for (int blk = 0; blk < NBLK; ++blk) {
    const float* Wq_b = Wq + (size_t)blk * DMODEL * DMODEL;
    const float* bq_b = bq + (size_t)blk * DMODEL;
    const float* Wk_b = Wk + (size_t)blk * DMODEL * DMODEL;
    const float* bk_b = bk + (size_t)blk * DMODEL;
    const float* Wv_b = Wv + (size_t)blk * DMODEL * DMODEL;
    const float* bv_b = bv + (size_t)blk * DMODEL;
    const float* nW1_b = nW1 + (size_t)blk * 2 * DMODEL * DHID;
    const float* nb1_b = nb1 + (size_t)blk * DHID;
    const float* nW2_b = nW2 + (size_t)blk * DHID * DMODEL;
    const float* nb2_b = nb2 + (size_t)blk * DMODEL;
    const float* eW1_b = eW1 + (size_t)blk * 65 * DHID;
    const float* eb1_b = eb1 + (size_t)blk * DHID;
    const float* eW2_b = eW2 + (size_t)blk * DHID;
    const float* eb2_b = eb2 + blk;

    auto intra = [&](float* x, int N) {
      gemm(x, nullptr, Wq_b, bq_b, nullptr, qb,  N, DMODEL, DMODEL, 0, 0);
      gemm(x, nullptr, Wk_b, bk_b, nullptr, kb,  N, DMODEL, DMODEL, 0, 0);
      gemm(x, nullptr, Wv_b, bv_b, nullptr, vvb, N, DMODEL, DMODEL, 0, 0);
      dim3 grid((N + 3) / 4, BATCH);
      attn_softmax_kernel<<<grid, 128, 0, stream>>>(qb, kb, vvb, aggb, N);
      gemm(x, aggb, nW1_b, nb1_b, nullptr, hid, N, DHID, 2 * DMODEL, DMODEL, 1);
      gemm(hid, nullptr, nW2_b, nb2_b, x, x, N, DMODEL, DHID, 0, 0);  // residual
    };

    auto inter = [&](float* x, int Nx, const float* y, int Ny,
                     const float* dis, float* e_out) {
      gemm(x, nullptr, Wq_b, bq_b, nullptr, qb,  Nx, DMODEL, DMODEL, 0, 0);  // xq
      gemm(y, nullptr, Wk_b, bk_b, nullptr, kb,  Ny, DMODEL, DMODEL, 0, 0);  // yk
      gemm(y, nullptr, Wv_b, bv_b, nullptr, vvb, Ny, DMODEL, DMODEL, 0, 0);  // yv
      gemm(qb, nullptr, eW1_b,             nullptr, nullptr, hqb, Nx, DHID, DMODEL, 0, 0);
      gemm(kb, nullptr, eW1_b + 32 * DHID, nullptr, nullptr, hkb, Ny, DHID, DMODEL, 0, 0);
      dim3 grid((Nx + 3) / 4, BATCH);
      inter_edge_kernel<<<grid, 128, 0, stream>>>(hqb, hkb, dis,
          eW1_b + 64 * DHID, eb1_b, eW2_b, eb2_b, vvb, aggb, e_out, Nx, Ny);
      gemm(x, aggb, nW1_b, nb1_b, nullptr, hid, Nx, DHID, 2 * DMODEL, DMODEL, 1);
      gemm(hid, nullptr, nW2_b, nb2_b, x, x, Nx, DMODEL, DHID, 0, 0);  // residual
    };

    intra(robot, NA);
    intra(frontier, NF);
    intra(rh, NH);
    intra(fh, NH);
    inter(robot,    NA, rh,       NH, dis_rp, nullptr);
    inter(frontier, NF, fh,       NH, dis_fp, nullptr);
    inter(robot,    NA, frontier, NF, dis_rf, (float*)d_out);  // edge -> d_out
  }
}